// TTrain_4526895530592
// MI455X (gfx1250) — compile-verified
//
#include <hip/hip_runtime.h>
#include <math.h>
#include <stdint.h>

// ---------------------------------------------------------------------------
// TTrain log|psi|^2 - lognorm for MI455X (gfx1250, wave32, WMMA, async-LDS).
//
// Part 1 (per-batch chains)  : sequential 64-matvec chain; gathered 16KB
//   matrix double-buffered in LDS via global_load_async_to_lds_b128 (index
//   stream is data-independent -> next matrix loads while current computes).
// Part 2 (model lognorm)     : persistent kernel, ~550 GFLOP of f32 64^3
//   matmuls tiled with v_wmma_f32_16x16x4_f32; triple-buffered transfer
//   matrix gives ONE grid barrier per step; tiles fetched with async-to-LDS.
// ---------------------------------------------------------------------------

#define D_BOND 64
#define D_PHYS 256
#define BATCH  128
#define SEQLEN 2048

#define LDSS   68               // padded LDS row stride (floats): 68*4 % 16 == 0
                                // and 68 % 64 == 4 -> conflict-free column reads
#define NWG2   128              // workgroups in persistent lognorm kernel
#define IPW    (D_PHYS / NWG2)  // phys indices per workgroup (= 2)

// d_ws float-offset layout (~49 KB)
#define WS_M0       0           // transfer-matrix ring buffer (3 x 64x64)
#define WS_M1       4096
#define WS_M2       8192
#define WS_LOGNORM  12288       // final scalar lognorm
#define WS_CNT      12290       // grid-barrier arrival counter (unsigned)
#define WS_GEN      12291       // grid-barrier generation       (unsigned)

typedef __attribute__((ext_vector_type(2))) float v2f;
typedef __attribute__((ext_vector_type(8))) float v8f;

// Wait until at most N async(global<->LDS) ops outstanding for this wave.
#define S_WAIT_ASYNC(N) asm volatile("s_wait_asynccnt %0" ::"i"(N) : "memory")

// ---------------------------------------------------------------------------
// Async 64x64 f32 tile fetch: global (row-major) -> padded LDS, no VGPR
// staging.  Per-lane LDS dest addresses let the async engine deposit straight
// into the stride-68 layout.  TPB threads issue 1024/TPB b128 ops each
// (=> 8*(128/TPB) ASYNCcnt increments per wave).
// ---------------------------------------------------------------------------
template <int TPB>
__device__ __forceinline__ void async_tile_ld(const float* __restrict__ g,
                                              float* lds) {
  const int tid = threadIdx.x;
#pragma unroll
  for (int r = 0; r < 1024 / TPB; ++r) {
    const int vi = tid + TPB * r;              // float4 index, 0..1023
    const int e = vi << 2;
    const int row = e >> 6, col = e & 63;      // col % 4 == 0 -> 16B aligned
    unsigned lo = (unsigned)(uintptr_t)&lds[row * LDSS + col];
    unsigned long long ga = (unsigned long long)(uintptr_t)(g + e);
    asm volatile("global_load_async_to_lds_b128 %0, %1, off"
                 ::"v"(lo), "v"(ga)
                 : "memory");
  }
}

// ---------------------------------------------------------------------------
// Sense-reversing grid barrier (NWG2 WGs of 128 threads, ~53KB LDS each:
// trivially co-resident on MI455X).
// ---------------------------------------------------------------------------
__device__ __forceinline__ void grid_sync(float* wsf) {
  unsigned* cnt = (unsigned*)(wsf + WS_CNT);
  unsigned* gen = (unsigned*)(wsf + WS_GEN);
  __syncthreads();
  if (threadIdx.x == 0) {
    __threadfence();
    unsigned g = __hip_atomic_load(gen, __ATOMIC_RELAXED, __HIP_MEMORY_SCOPE_AGENT);
    unsigned a = __hip_atomic_fetch_add(cnt, 1u, __ATOMIC_ACQ_REL, __HIP_MEMORY_SCOPE_AGENT);
    if (a == (unsigned)(NWG2 - 1)) {
      __hip_atomic_store(cnt, 0u, __ATOMIC_RELAXED, __HIP_MEMORY_SCOPE_AGENT);
      __hip_atomic_fetch_add(gen, 1u, __ATOMIC_RELEASE, __HIP_MEMORY_SCOPE_AGENT);
    } else {
      while (__hip_atomic_load(gen, __ATOMIC_ACQUIRE, __HIP_MEMORY_SCOPE_AGENT) == g) {
        __builtin_amdgcn_s_sleep(2);
      }
    }
    __threadfence();
  }
  __syncthreads();
}

// ---------------------------------------------------------------------------
// One contribution  macc += C^T * U * C  (all 64x64, f32) via WMMA 16x16x4.
// 4 waves per WG; wave w owns output tile-row w.  Caller guarantees ldsC/ldsU
// are ready and synchronizes around the call.
// ---------------------------------------------------------------------------
__device__ __forceinline__ void gemm_CtUC(const float* __restrict__ ldsC,
                                          const float* __restrict__ ldsU,
                                          float* __restrict__ ldsT, v8f macc[4],
                                          const int wave, const int hi,
                                          const int lr) {
  // GEMM1: T[k,l] = sum_j C[j,k] * U[j,l]   (A = C^T, B = U)
  v8f tacc[4];
#pragma unroll
  for (int lT = 0; lT < 4; ++lT)
    for (int g = 0; g < 8; ++g) tacc[lT][g] = 0.0f;
#pragma unroll
  for (int jc = 0; jc < 16; ++jc) {
    const int j0 = jc * 4 + 2 * hi;
    v2f a;                                   // A frag: lanes=M(k), vgprs=K(j)
    a.x = ldsC[(j0    ) * LDSS + wave * 16 + lr];
    a.y = ldsC[(j0 + 1) * LDSS + wave * 16 + lr];
#pragma unroll
    for (int lT = 0; lT < 4; ++lT) {
      v2f b;                                 // B frag: lanes=N(l), vgprs=K(j)
      b.x = ldsU[(j0    ) * LDSS + lT * 16 + lr];
      b.y = ldsU[(j0 + 1) * LDSS + lT * 16 + lr];
      tacc[lT] = __builtin_amdgcn_wmma_f32_16x16x4_f32(
          false, a, false, b, (short)0, tacc[lT], false, false);
    }
  }
#pragma unroll
  for (int lT = 0; lT < 4; ++lT)
    for (int g = 0; g < 8; ++g)
      ldsT[(wave * 16 + g + 8 * hi) * LDSS + lT * 16 + lr] = tacc[lT][g];
  __syncthreads();

  // GEMM2: macc[k,m] += sum_l T[k,l] * C[l,m]
#pragma unroll
  for (int lc = 0; lc < 16; ++lc) {
    const int l0 = lc * 4 + 2 * hi;
    v2f a;                 // column read of T: stride-68 -> conflict-free
    a.x = ldsT[(wave * 16 + lr) * LDSS + l0    ];
    a.y = ldsT[(wave * 16 + lr) * LDSS + l0 + 1];
#pragma unroll
    for (int mT = 0; mT < 4; ++mT) {
      v2f b;
      b.x = ldsC[(l0    ) * LDSS + mT * 16 + lr];
      b.y = ldsC[(l0 + 1) * LDSS + mT * 16 + lr];
      macc[mT] = __builtin_amdgcn_wmma_f32_16x16x4_f32(
          false, a, false, b, (short)0, macc[mT], false, false);
    }
  }
}

// ---------------------------------------------------------------------------
// Part 2: lognorm.  Step: M' = sum_i core_i^T U core_i, inf-norm
// renormalized in log space, 2047 steps.  Triple-buffered M ring:
//   step s: read M[(s-1)%3] (normalize on the fly), accumulate M[s%3],
//           zero M[(s+1)%3]  -> single grid barrier per step.
// ---------------------------------------------------------------------------
__global__ __launch_bounds__(128) void tt_lognorm_kernel(
    const float* __restrict__ core, const float* __restrict__ lb,
    const float* __restrict__ rb, float* __restrict__ ws) {
  __shared__ float ldsC[D_BOND * LDSS];   // core_i tile
  __shared__ float ldsU[D_BOND * LDSS];   // U tile (raw M, rescaled in place)
  __shared__ float ldsT[D_BOND * LDSS];   // T = C^T U intermediate
  __shared__ float ldsU0[64];             // left boundary vector u0
  __shared__ float ldsV[64];              // reductions
  __shared__ float ldsS[2];

  float* Mb0 = ws + WS_M0;
  float* Mb1 = ws + WS_M1;
  float* Mb2 = ws + WS_M2;

  const int tid  = threadIdx.x;
  const int wg   = blockIdx.x;
  const int wave = tid >> 5;
  const int lane = tid & 31;
  const int hi   = lane >> 4;
  const int lr   = lane & 15;

  float acc_log = 0.0f;                   // valid on (wg==0, tid==0)

  // ---- preamble: u0 = lb/max|lb|; zero M0,M2; acc = log Z0 --------------
  if (tid < 64) ldsU0[tid] = lb[tid];
  __syncthreads();
  if (tid == 0) {
    float z = 0.0f;
    for (int j = 0; j < 64; ++j) z = fmaxf(z, fabsf(ldsU0[j]));
    ldsS[0] = z;
  }
  __syncthreads();
  const float Z0 = ldsS[0];
  if (tid < 64) ldsU0[tid] = ldsU0[tid] / Z0;
  if (wg == 0 && tid == 0) acc_log = logf(Z0);
  if (tid < 32) {                         // 128 WGs x 32 = 4096 each
    Mb0[wg * 32 + tid] = 0.0f;
    Mb2[wg * 32 + tid] = 0.0f;
  }
  grid_sync(ws);

  // ---- first site: M1 = A^T A (into Mb2), A[i,k] = sum_j u0[j] C_i[j,k] --
  for (int ii = 0; ii < IPW; ++ii) {
    const int i = wg * IPW + ii;
    async_tile_ld<128>(core + i * 4096, ldsC);
    S_WAIT_ASYNC(0);
    __syncthreads();
    if (tid < 64) {
      float a = 0.0f;
      for (int j = 0; j < 64; ++j) a = fmaf(ldsU0[j], ldsC[j * LDSS + tid], a);
      ldsV[tid] = a;
    }
    __syncthreads();
    for (int p = tid; p < 4096; p += 128)
      atomicAdd(&Mb2[p], ldsV[p >> 6] * ldsV[p & 63]);
    __syncthreads();
  }
  grid_sync(ws);

  // ---- main chain: 2047 steps, 1 grid barrier each ----------------------
  float* Mprev = Mb2;
  float* Mcur  = Mb0;
  float* Mnxt  = Mb1;
  const int i0 = wg * IPW;
  const int i1 = wg * IPW + 1;

  for (int s = 0; s < SEQLEN - 1; ++s) {
    // issue async fetches: raw M_prev and first core tile
    async_tile_ld<128>(Mprev, ldsU);              // 8 per wave
    async_tile_ld<128>(core + i0 * 4096, ldsC);   // +8 per wave
    S_WAIT_ASYNC(8);                              // U done (in-order), C flying
    __syncthreads();

    // redundant-per-WG normalize prep (overlaps the C tile flight):
    // Z = inf-norm(M_prev); ldsU *= 1/Z; zero my slice of M_next
    if (tid < 64) {
      float sm = 0.0f;
      for (int m = 0; m < 64; ++m) sm += fabsf(ldsU[tid * LDSS + m]);
      ldsV[tid] = sm;
    }
    __syncthreads();
    if (tid == 0) {
      float z = 0.0f;
      for (int j = 0; j < 64; ++j) z = fmaxf(z, ldsV[j]);
      ldsS[0] = 1.0f / z;
      ldsS[1] = logf(z);
    }
    __syncthreads();
    const float invz = ldsS[0];
    if (wg == 0 && tid == 0) acc_log += ldsS[1];
#pragma unroll
    for (int r = 0; r < 8; ++r) {
      const int vi = tid + 128 * r;
      const int e = vi << 2;
      float4* p = (float4*)&ldsU[(e >> 6) * LDSS + (e & 63)];
      float4 v = *p;
      v.x *= invz; v.y *= invz; v.z *= invz; v.w *= invz;
      *p = v;
    }
    if (tid < 32) Mnxt[wg * 32 + tid] = 0.0f;

    v8f macc[4];
#pragma unroll
    for (int mT = 0; mT < 4; ++mT)
      for (int g = 0; g < 8; ++g) macc[mT][g] = 0.0f;

    S_WAIT_ASYNC(0);                 // C tile landed
    __syncthreads();                 // + rescaled U published
    gemm_CtUC(ldsC, ldsU, ldsT, macc, wave, hi, lr);
    __syncthreads();                 // everyone done reading ldsC/ldsT

    async_tile_ld<128>(core + i1 * 4096, ldsC);
    S_WAIT_ASYNC(0);
    __syncthreads();
    gemm_CtUC(ldsC, ldsU, ldsT, macc, wave, hi, lr);

    // accumulate partial M' (4096 f32 atomics per WG per step)
#pragma unroll
    for (int mT = 0; mT < 4; ++mT)
      for (int g = 0; g < 8; ++g)
        atomicAdd(&Mcur[(wave * 16 + g + 8 * hi) * 64 + mT * 16 + lr],
                  macc[mT][g]);

    grid_sync(ws);
    float* t = Mprev; Mprev = Mcur; Mcur = Mnxt; Mnxt = t;
  }

  // ---- finish: final Z cancels ->  lognorm = acc + log|rb^T M rb| -------
  if (wg == 0) {
    if (tid < 64) {
      float inner = 0.0f;
      for (int m = 0; m < 64; ++m)
        inner = fmaf(Mprev[tid * 64 + m], rb[m], inner);
      ldsV[tid] = rb[tid] * inner;
    }
    __syncthreads();
    if (tid == 0) {
      float o = 0.0f;
      for (int j = 0; j < 64; ++j) o += ldsV[j];
      ws[WS_LOGNORM] = acc_log + logf(fabsf(o));
    }
  }
}

// ---------------------------------------------------------------------------
// Part 1: per-batch chains.  One WG (64 threads, 2 waves) per batch element.
// The gathered 16KB matrix for step t+1 streams into the spare LDS buffer
// via async-to-LDS while step t computes from the other -> L2 gather latency
// fully hidden.  Thread j owns output column j (stride-68 LDS: conflict-free).
// Stream-ordered after tt_lognorm_kernel, so it reads the lognorm scalar.
// ---------------------------------------------------------------------------
__global__ __launch_bounds__(64) void tt_logpsi_kernel(
    const int* __restrict__ x, const float* __restrict__ core,
    const float* __restrict__ lb, const float* __restrict__ rb,
    const float* __restrict__ ws, float* __restrict__ out) {
  __shared__ float ldsW[2][D_BOND * LDSS];
  __shared__ float u[64];
  __shared__ float wred[2];
  const int b = blockIdx.x;
  const int j = threadIdx.x;
  const int w = j >> 5;

  // Z0 = max|lb|; u = lb/Z0; acc = log Z0
  float m0 = fabsf(lb[j]);
  for (int off = 16; off; off >>= 1) m0 = fmaxf(m0, __shfl_xor(m0, off, 32));
  if ((j & 31) == 0) wred[w] = m0;
  __syncthreads();
  float Z = fmaxf(wred[0], wred[1]);
  u[j] = lb[j] / Z;
  float acc = logf(Z);
  __syncthreads();

  const int* xb = x + b * SEQLEN;

  // preload matrix for t = 0
  async_tile_ld<64>(core + xb[0] * 4096, ldsW[0]);
  S_WAIT_ASYNC(0);
  __syncthreads();

  for (int t = 0; t < SEQLEN; ++t) {
    if (t + 1 < SEQLEN)   // data-independent index stream: prefetch to LDS
      async_tile_ld<64>(core + xb[t + 1] * 4096, ldsW[(t + 1) & 1]);

    const float* W = ldsW[t & 1];
    float vj = 0.0f;
#pragma unroll 16
    for (int i = 0; i < 64; ++i) vj = fmaf(u[i], W[i * LDSS + j], vj);

    float mv = fabsf(vj);
    for (int off = 16; off; off >>= 1) mv = fmaxf(mv, __shfl_xor(mv, off, 32));
    if ((j & 31) == 0) wred[w] = mv;
    __syncthreads();                 // also: everyone done reading u[]
    Z = fmaxf(wred[0], wred[1]);
    u[j] = vj / Z;
    acc += logf(Z);
    __syncthreads();

    S_WAIT_ASYNC(0);                 // next-step matrix landed
    __syncthreads();
  }

  float dj = u[j] * rb[j];
  for (int off = 16; off; off >>= 1) dj += __shfl_xor(dj, off, 32);
  if ((j & 31) == 0) wred[w] = dj;
  __syncthreads();
  if (j == 0) {
    float dot = wred[0] + wred[1];
    out[b] = 2.0f * (acc + logf(fabsf(dot))) - ws[WS_LOGNORM];
  }
}

__global__ void tt_init_kernel(float* ws) {
  ((unsigned*)(ws + WS_CNT))[0] = 0u;
  ((unsigned*)(ws + WS_GEN))[0] = 0u;
  ws[WS_LOGNORM] = 0.0f;
}

extern "C" void kernel_launch(void* const* d_in, const int* in_sizes, int n_in,
                              void* d_out, int out_size, void* d_ws,
                              size_t ws_size, hipStream_t stream) {
  const int*   x    = (const int*)d_in[0];     // (128, 2048) int32
  const float* core = (const float*)d_in[1];   // (256, 64, 64) f32
  const float* lb   = (const float*)d_in[2];   // (64,) f32
  const float* rb   = (const float*)d_in[3];   // (64,) f32
  float* out = (float*)d_out;                  // (128,) f32
  float* ws  = (float*)d_ws;

  tt_init_kernel<<<1, 1, 0, stream>>>(ws);
  tt_lognorm_kernel<<<NWG2, 128, 0, stream>>>(core, lb, rb, ws);
  tt_logpsi_kernel<<<BATCH, 64, 0, stream>>>(x, core, lb, rb, ws, out);
}